// HierarchicalModel_74586402062651
// MI455X (gfx1250) — compile-verified
//
#include <hip/hip_runtime.h>

// ---------------------------------------------------------------------------
// PhysNet pipeline for MI455X (gfx1250, wave32, WMMA + Tensor Data Mover).
// GEMMs on the matrix pipe via V_WMMA_F32_16X16X4_F32 (fp32-exact);
// LDS staging of weight/radial tiles via TENSOR_LOAD_TO_LDS (TENSORcnt).
// ---------------------------------------------------------------------------

typedef float v2f __attribute__((ext_vector_type(2)));
typedef float v8f __attribute__((ext_vector_type(8)));
typedef unsigned int u32x4 __attribute__((ext_vector_type(4)));
typedef int i32x4 __attribute__((ext_vector_type(4)));
typedef int i32x8 __attribute__((ext_vector_type(8)));

#define FDIM 128
#define RDIM 64

__device__ __forceinline__ float ssp_f(float x) {
  // softplus(x) - ln2, numerically stable
  return fmaxf(x, 0.0f) + __logf(1.0f + __expf(-fabsf(x))) - 0.69314718055994531f;
}

__device__ __forceinline__ v8f wmma_f32(v2f a, v2f b, v8f c) {
  return __builtin_amdgcn_wmma_f32_16x16x4_f32(false, a, false, b, (short)0, c,
                                               false, false);
}

// TDM: 1-D contiguous copy of nelem f32 words from global -> LDS, with
// optional LDS padding (pad ctrl bits per D# group1 [20],[24:22],[31:25]).
// D# group0: count=1, lds_addr, 57-bit global addr, type=2.
// D# group1: data_size=4B, tensor_dim0=tile_dim0=nelem, dim1=1.
__device__ __forceinline__ void tdm_load_1d(unsigned lds_addr, const void* gsrc,
                                            unsigned nelem, unsigned padbits) {
  const unsigned long long ga = (unsigned long long)gsrc;
  u32x4 g0;
  g0[0] = 1u;                                                  // count=1
  g0[1] = lds_addr;                                            // LDS byte addr
  g0[2] = (unsigned)ga;                                        // gaddr lo
  g0[3] = ((unsigned)(ga >> 32) & 0x01FFFFFFu) | (2u << 30);   // gaddr hi|type=2
  i32x8 g1;
  g1[0] = (int)((2u << 16) | padbits);          // data_size=4B + pad control
  g1[1] = (int)((nelem & 0xFFFFu) << 16);       // tensor_dim0[15:0]
  g1[2] = (int)(((nelem >> 16) & 0xFFFFu) | (1u << 16));  // dim0 hi | dim1=1
  g1[3] = (int)((nelem & 0xFFFFu) << 16);       // dim1 hi(0) | tile_dim0
  g1[4] = 1;                                    // tile_dim1=1, tile_dim2=0
  g1[5] = (int)nelem;                           // tensor_dim0_stride lo
  g1[6] = 0;
  g1[7] = 0;
  i32x4 z4 = {0, 0, 0, 0};
#if __clang_major__ >= 23
  i32x8 z8 = {0, 0, 0, 0, 0, 0, 0, 0};
  __builtin_amdgcn_tensor_load_to_lds(g0, g1, z4, z4, z8, 0);
#else
  __builtin_amdgcn_tensor_load_to_lds(g0, g1, z4, z4, 0);
#endif
}

// pad 4 DWORDs after every 64 DWORDs stored -> LDS row stride 68
#define RS_PADBITS ((1u << 20) | (5u << 22) | (3u << 25))

// ---------------------------------------------------------------------------
// Interleave a KxF row-major weight into (k,k+1) float2 pairs so TDM can
// stage it as a flat copy while frags stay single ds_load_b64.
// out[(k>>1)*128+col] = {W[k][col], W[k+1][col]} ; n = (K/2)*128
// ---------------------------------------------------------------------------
__global__ __launch_bounds__(256) void interleave_kernel(
    const float* __restrict__ W, float2* __restrict__ out, int n) {
  const int i = blockIdx.x * blockDim.x + threadIdx.x;
  if (i < n) {
    const int k2 = i >> 7, col = i & 127;
    out[i] = make_float2(W[(2 * k2) * FDIM + col], W[(2 * k2 + 1) * FDIM + col]);
  }
}

// ---------------------------------------------------------------------------
// Dense GEMM:  Y[16-row tiles] = epilogue( act(X) @ W + b )
// One wave computes a full 16x128 output tile; 4 waves/WG; W staged via TDM.
// ---------------------------------------------------------------------------
template <bool IN_SSP, bool OUT_SSP, int ADDMODE>
__global__ __launch_bounds__(128) void gemm128_kernel(
    const float* __restrict__ X, const float2* __restrict__ WLg,
    const float* __restrict__ bias, const float* __restrict__ resid,
    const float* __restrict__ gvec, float* __restrict__ Y) {
  __shared__ float2 WL[64 * FDIM];      // 64 KB interleaved weight
  __shared__ float AS[4][16][132];      // per-wave activated input tile
  __shared__ float bL[FDIM];
  __shared__ float gL[FDIM];

  const int tid = threadIdx.x;
  const int wave = tid >> 5;
  const int lane = tid & 31;
  const int row = lane & 15;
  const int half = lane >> 4;

  if (wave == 0) {
    tdm_load_1d((unsigned)(unsigned long long)&WL[0], WLg, 64 * FDIM * 2, 0u);
    __builtin_amdgcn_s_wait_tensorcnt(0);
  }
  bL[tid] = bias[tid];
  if (ADDMODE == 2) gL[tid] = gvec[tid];

  const int rowBase = (blockIdx.x * 4 + wave) * 16;
  for (int r = 0; r < 16; ++r) {
    float4 v = *(const float4*)(X + (size_t)(rowBase + r) * FDIM + lane * 4);
    if (IN_SSP) { v.x = ssp_f(v.x); v.y = ssp_f(v.y); v.z = ssp_f(v.z); v.w = ssp_f(v.w); }
    *(float4*)&AS[wave][r][lane * 4] = v;
  }
  __syncthreads();

  for (int nt = 0; nt < 8; ++nt) {
    const int col = nt * 16 + row;
    const float bb = bL[col];
    v8f acc = {bb, bb, bb, bb, bb, bb, bb, bb};
#pragma unroll 8
    for (int k0 = 0; k0 < FDIM; k0 += 4) {
      const int k = k0 + 2 * half;
      v2f a = *(const v2f*)&AS[wave][row][k];
      const float2 bw = WL[(k >> 1) * FDIM + col];
      v2f b; b.x = bw.x; b.y = bw.y;
      acc = wmma_f32(a, b, acc);
    }
#pragma unroll
    for (int r = 0; r < 8; ++r) {
      const int gRow = rowBase + r + 8 * half;
      float val = acc[r];
      if (OUT_SSP) val = ssp_f(val);
      if (ADDMODE == 1) val += resid[(size_t)gRow * FDIM + col];
      if (ADDMODE == 2) val += resid[(size_t)gRow * FDIM + col] * gL[col];
      Y[(size_t)gRow * FDIM + col] = val;
    }
  }
}

// ---------------------------------------------------------------------------
// Fused pair kernel, grid-stride over 16-pair tiles (weights staged once/WG):
//   gate = radial_tile @ W_gate            (WMMA frags; radial via padded TDM)
//   per side: fm = ssp(sfeat[idx] @ W_J + bJ); atomicAdd(proto, gate*fm)
// ---------------------------------------------------------------------------
__global__ __launch_bounds__(128) void pair_kernel(
    const float* __restrict__ sfeat, const float* __restrict__ radial,
    const int* __restrict__ idx12, const float2* __restrict__ WJg,
    const float* __restrict__ bJ, const float2* __restrict__ Wgg,
    float* __restrict__ proto, int P) {
  __shared__ float2 WJL[64 * FDIM];   // 64 KB
  __shared__ float2 WGL[32 * FDIM];   // 32 KB
  __shared__ float bJL[FDIM];
  __shared__ float AS[4][16][132];    // gathered feature tiles
  __shared__ float RS[4][16][68];     // radial tiles (stride 68 via TDM pad)
  __shared__ int IL[4][2][16];

  const int tid = threadIdx.x;
  const int wave = tid >> 5;
  const int lane = tid & 31;
  const int row = lane & 15;
  const int half = lane >> 4;

  if (wave == 0) {
    tdm_load_1d((unsigned)(unsigned long long)&WJL[0], WJg, 64 * FDIM * 2, 0u);
    __builtin_amdgcn_s_wait_tensorcnt(0);
  } else if (wave == 1) {
    tdm_load_1d((unsigned)(unsigned long long)&WGL[0], Wgg, 32 * FDIM * 2, 0u);
    __builtin_amdgcn_s_wait_tensorcnt(0);
  }
  bJL[tid] = bJ[tid];
  __syncthreads();

  const int numTiles = P >> 4;
  const int waveId = blockIdx.x * 4 + wave;
  const int nWaves = gridDim.x * 4;
  const unsigned rsAddr = (unsigned)(unsigned long long)&RS[wave][0][0];

  for (int t = waveId; t < numTiles; t += nWaves) {
    const int p0 = t * 16;
    // async: radial tile -> LDS with stride-68 padding
    tdm_load_1d(rsAddr, radial + (size_t)p0 * RDIM, 16 * RDIM, RS_PADBITS);
    // pair indices: lanes 0-15 -> side 0, lanes 16-31 -> side 1
    IL[wave][half][row] = idx12[half * P + p0 + row];
    __builtin_amdgcn_s_wait_tensorcnt(0);

    // gate fragments (no bias)
    v8f g[8];
#pragma unroll
    for (int nt = 0; nt < 8; ++nt) {
      const int col = nt * 16 + row;
      v8f acc = {0.f, 0.f, 0.f, 0.f, 0.f, 0.f, 0.f, 0.f};
#pragma unroll
      for (int k0 = 0; k0 < RDIM; k0 += 4) {
        const int k = k0 + 2 * half;
        v2f a = *(const v2f*)&RS[wave][row][k];
        const float2 bw = WGL[(k >> 1) * FDIM + col];
        v2f b; b.x = bw.x; b.y = bw.y;
        acc = wmma_f32(a, b, acc);
      }
      g[nt] = acc;
    }

    for (int side = 0; side < 2; ++side) {
      for (int r = 0; r < 16; ++r) {
        const int arow = IL[wave][side][r];
        float4 v = *(const float4*)(sfeat + (size_t)arow * FDIM + lane * 4);
        *(float4*)&AS[wave][r][lane * 4] = v;
      }
      int prow[8];
#pragma unroll
      for (int r = 0; r < 8; ++r) prow[r] = IL[wave][side][r + 8 * half];

      for (int nt = 0; nt < 8; ++nt) {
        const int col = nt * 16 + row;
        const float bb = bJL[col];
        v8f acc = {bb, bb, bb, bb, bb, bb, bb, bb};
#pragma unroll 8
        for (int k0 = 0; k0 < FDIM; k0 += 4) {
          const int k = k0 + 2 * half;
          v2f a = *(const v2f*)&AS[wave][row][k];
          const float2 bw = WJL[(k >> 1) * FDIM + col];
          v2f b; b.x = bw.x; b.y = bw.y;
          acc = wmma_f32(a, b, acc);
        }
        const v8f gg = g[nt];
#pragma unroll
        for (int r = 0; r < 8; ++r) {
          const float val = ssp_f(acc[r]) * gg[r];
          atomicAdd(&proto[(size_t)prow[r] * FDIM + col], val);
        }
      }
    }
  }
}

// ---------------------------------------------------------------------------
// energies[row] += ssp(T[row]) . W_out + b_out   (wave-per-row GEMV)
// ---------------------------------------------------------------------------
__global__ __launch_bounds__(256) void energy_kernel(
    const float* __restrict__ T, const float* __restrict__ Wout,
    const float* __restrict__ bout, float* __restrict__ energies, int nRows) {
  const int lane = threadIdx.x & 31;
  const int gw = blockIdx.x * (blockDim.x >> 5) + (threadIdx.x >> 5);
  const int nw = gridDim.x * (blockDim.x >> 5);
  const float4 w = *(const float4*)(Wout + lane * 4);
  const float b = bout[0];
  for (int row = gw; row < nRows; row += nw) {
    const float4 t = *(const float4*)(T + (size_t)row * FDIM + lane * 4);
    float p = ssp_f(t.x) * w.x + ssp_f(t.y) * w.y + ssp_f(t.z) * w.z + ssp_f(t.w) * w.w;
    for (int off = 16; off; off >>= 1) p += __shfl_xor(p, off, 32);
    if (lane == 0) energies[row] += p + b;
  }
}

__global__ __launch_bounds__(256) void ssp_vec_kernel(
    const float* __restrict__ in, float* __restrict__ out, int n4) {
  for (int i = blockIdx.x * blockDim.x + threadIdx.x; i < n4;
       i += gridDim.x * blockDim.x) {
    float4 v = ((const float4*)in)[i];
    v.x = ssp_f(v.x); v.y = ssp_f(v.y); v.z = ssp_f(v.z); v.w = ssp_f(v.w);
    ((float4*)out)[i] = v;
  }
}

__global__ __launch_bounds__(256) void zero_kernel(float* __restrict__ p, int n) {
  const int i = blockIdx.x * blockDim.x + threadIdx.x;
  if (i < n) p[i] = 0.0f;
}

// ---------------------------------------------------------------------------
extern "C" void kernel_launch(void* const* d_in, const int* in_sizes, int n_in,
                              void* d_out, int out_size, void* d_ws,
                              size_t ws_size, hipStream_t stream) {
  const int F = FDIM;
  const int N = in_sizes[1] / F;            // 8192
  const int P = in_sizes[3] / 2;            // 400000
  const int L = in_sizes[4] / (F * F);      // 5

  const float* features = (const float*)d_in[1];
  const float* radial   = (const float*)d_in[2];
  const int*   idx12    = (const int*)d_in[3];
  const float* W_I   = (const float*)d_in[4];
  const float* b_I   = (const float*)d_in[5];
  const float* W_J   = (const float*)d_in[6];
  const float* b_J   = (const float*)d_in[7];
  const float* W_g   = (const float*)d_in[8];
  const float* gvec  = (const float*)d_in[9];
  const float* W_int = (const float*)d_in[10];
  const float* b_int = (const float*)d_in[11];
  const float* ri_W1 = (const float*)d_in[12];
  const float* ri_b1 = (const float*)d_in[13];
  const float* ri_W2 = (const float*)d_in[14];
  const float* ri_b2 = (const float*)d_in[15];
  const float* ra_W1 = (const float*)d_in[16];
  const float* ra_b1 = (const float*)d_in[17];
  const float* ra_W2 = (const float*)d_in[18];
  const float* ra_b2 = (const float*)d_in[19];
  const float* ro_W1 = (const float*)d_in[20];
  const float* ro_b1 = (const float*)d_in[21];
  const float* ro_W2 = (const float*)d_in[22];
  const float* ro_b2 = (const float*)d_in[23];
  const float* W_out = (const float*)d_in[24];
  const float* b_out = (const float*)d_in[25];

  const size_t NF = (size_t)N * F;
  float* ws    = (float*)d_ws;
  float* featA = ws;
  float* featB = ws + NF;
  float* sfeat = ws + 2 * NF;
  float* X     = ws + 3 * NF;   // proto / msg / ro-chain output
  float* H     = ws + 4 * NF;   // residual-block hidden
  float2* wbuf = (float2*)(ws + 5 * NF);  // interleaved weights, 16 slots/module

  auto wslot = [&](int l, int s) -> float2* {
    return wbuf + ((size_t)l * 16 + s) * (size_t)(64 * F);
  };
  // slots: 0=W_I 1=W_J 2=W_gate 3=W_int 4..6=ri1 7..9=ri2 10..11=ra1
  //        12..13=ra2 14=ro1 15=ro2
  for (int l = 0; l < L; ++l) {
    const size_t FF = (size_t)F * F;
    auto prep = [&](const float* src, float2* dst, int K) {
      const int n = (K / 2) * F;
      interleave_kernel<<<(n + 255) / 256, 256, 0, stream>>>(src, dst, n);
    };
    prep(W_I + l * FF, wslot(l, 0), F);
    prep(W_J + l * FF, wslot(l, 1), F);
    prep(W_g + (size_t)l * RDIM * F, wslot(l, 2), RDIM);
    prep(W_int + l * FF, wslot(l, 3), F);
    for (int j = 0; j < 3; ++j) {
      prep(ri_W1 + ((size_t)l * 3 + j) * FF, wslot(l, 4 + j), F);
      prep(ri_W2 + ((size_t)l * 3 + j) * FF, wslot(l, 7 + j), F);
    }
    for (int j = 0; j < 2; ++j) {
      prep(ra_W1 + ((size_t)l * 2 + j) * FF, wslot(l, 10 + j), F);
      prep(ra_W2 + ((size_t)l * 2 + j) * FF, wslot(l, 12 + j), F);
    }
    prep(ro_W1 + l * FF, wslot(l, 14), F);
    prep(ro_W2 + l * FF, wslot(l, 15), F);
  }

  float* energies = (float*)d_out;       // [N]
  float* feat_out = (float*)d_out + N;   // [N,F]

  const int gemmGrid = N / 64;           // 4 waves x 16 rows per WG
  const int pairGrid = 512;              // grid-stride over P/16 tiles

  zero_kernel<<<(N + 255) / 256, 256, 0, stream>>>(energies, N);
  hipMemcpyAsync(featA, features, NF * sizeof(float), hipMemcpyDeviceToDevice,
                 stream);

  float* fcur = featA;
  float* fnext = featB;
  for (int l = 0; l < L; ++l) {
    // sfeat = ssp(feat)
    ssp_vec_kernel<<<512, 256, 0, stream>>>(fcur, sfeat, (int)(NF / 4));
    // X = h_self = ssp(sfeat @ W_I + b_I)   (scatter target pre-initialized)
    gemm128_kernel<false, true, 0><<<gemmGrid, 128, 0, stream>>>(
        sfeat, wslot(l, 0), b_I + (size_t)l * F, nullptr, nullptr, X);
    // X += scatter of gate * ssp(sfeat[idx] @ W_J + b_J)
    pair_kernel<<<pairGrid, 128, 0, stream>>>(
        sfeat, radial, idx12, wslot(l, 1), b_J + (size_t)l * F, wslot(l, 2), X, P);
    // ri residual chain (3 blocks)
    for (int j = 0; j < 3; ++j) {
      const size_t bo = ((size_t)l * 3 + j) * F;
      gemm128_kernel<true, false, 0><<<gemmGrid, 128, 0, stream>>>(
          X, wslot(l, 4 + j), ri_b1 + bo, nullptr, nullptr, H);
      gemm128_kernel<true, false, 1><<<gemmGrid, 128, 0, stream>>>(
          H, wslot(l, 7 + j), ri_b2 + bo, X, nullptr, X);
    }
    // feat2 = feat * gvec + ssp(msg) @ W_int + b_int
    gemm128_kernel<true, false, 2><<<gemmGrid, 128, 0, stream>>>(
        X, wslot(l, 3), b_int + (size_t)l * F, fcur, gvec + (size_t)l * F, fnext);
    // ra residual chain (2 blocks) in-place on fnext
    for (int j = 0; j < 2; ++j) {
      const size_t bo = ((size_t)l * 2 + j) * F;
      gemm128_kernel<true, false, 0><<<gemmGrid, 128, 0, stream>>>(
          fnext, wslot(l, 10 + j), ra_b1 + bo, nullptr, nullptr, H);
      gemm128_kernel<true, false, 1><<<gemmGrid, 128, 0, stream>>>(
          H, wslot(l, 12 + j), ra_b2 + bo, fnext, nullptr, fnext);
    }
    // ro residual chain (1 block) -> X  (feat2 itself is NOT modified)
    {
      const size_t bo = (size_t)l * F;
      gemm128_kernel<true, false, 0><<<gemmGrid, 128, 0, stream>>>(
          fnext, wslot(l, 14), ro_b1 + bo, nullptr, nullptr, H);
      gemm128_kernel<true, false, 1><<<gemmGrid, 128, 0, stream>>>(
          H, wslot(l, 15), ro_b2 + bo, fnext, nullptr, X);
    }
    // energies += ssp(X) @ W_out + b_out
    energy_kernel<<<32, 256, 0, stream>>>(X, W_out + (size_t)l * F, b_out + l,
                                          energies, N);
    float* t = fcur; fcur = fnext; fnext = t;
  }
  hipMemcpyAsync(feat_out, fcur, NF * sizeof(float), hipMemcpyDeviceToDevice,
                 stream);
}